// STGN_LSTM_22196390986555
// MI455X (gfx1250) — compile-verified
//
#include <hip/hip_runtime.h>
#include <hip/hip_bf16.h>
#include <math.h>

// ---------------------------------------------------------------------------
// STGN-LSTM, degenerate form (h0/c0 == 0, recurrence unused, softmax over a
// single token == identity).  Only X_seq[:, T-1, :] matters.
//   B=4096, T=256, D=64 (+2 delta cols -> 66 features), H=128
// ---------------------------------------------------------------------------

#define BATCH 4096
#define TLEN  256
#define DIN   64
#define FEAT  66
#define HDIM  128

typedef float v2f __attribute__((ext_vector_type(2)));
typedef float v8f __attribute__((ext_vector_type(8)));

__device__ __forceinline__ v8f wmma_f32(v2f a, v2f b, v8f c) {
    // D(16x16,f32) = A(16x4,f32) x B(4x16,f32) + C
    return __builtin_amdgcn_wmma_f32_16x16x4_f32(
        /*neg_a=*/false, a, /*neg_b=*/false, b,
        /*c_mod=*/(short)0, c, /*reuse_a=*/false, /*reuse_b=*/false);
}

__device__ __forceinline__ float sigm(float x) { return 1.0f / (1.0f + __expf(-x)); }

// ---------------------------------------------------------------------------
// Workspace layout (floats):
//   [   0..127 ] biEff   = Wih_b + Wix_b + bi
//   [ 128..255 ] bTEff   = WTh_b + WTx_b + WTt_b + bT
//   [ 256..383 ] bLEff   = WLh_b + WLx_b + WLdelta_b + bL
//   [ 384..511 ] bzEff   = WzT_b + WzL_b
//   [ 512..639 ] buEff   = Wuh_b + Wux_b + bu
//   [ 640..767 ] boEff   = Woh_b + Wox_b + Woz_b + bo
//   [ 768..895 ] brEff   = Wr_b + br
//   [ 896..1023] bhEff   = Wih_final_b + Ws_final_b + Wr_final_b + bh_final
//   [ 1024     ] wyEff   = Wy_b[0] + by[0]
//   [ 2048..18431 ] Wfin = Wih_final_w + Ws_final_w   (128x128 row-major)
// ---------------------------------------------------------------------------

__global__ __launch_bounds__(256) void stgn_prep(
    const float* Wih_b, const float* Wix_b, const float* bi,
    const float* WTh_b, const float* WTx_b, const float* WTt_b, const float* bT,
    const float* WLh_b, const float* WLx_b, const float* WLd_b, const float* bL,
    const float* WzT_b, const float* WzL_b,
    const float* Wuh_b, const float* Wux_b, const float* bu,
    const float* Woh_b, const float* Wox_b, const float* Woz_b, const float* bo,
    const float* Wr_b,  const float* br,
    const float* Wihf_b, const float* Wsf_b, const float* Wrf_b, const float* bhf,
    const float* Wy_b,  const float* by,
    const float* Wihf_w, const float* Wsf_w,
    float* ws)
{
    int t = blockIdx.x * 256 + threadIdx.x;
    if (t < HDIM * HDIM) {
        ws[2048 + t] = Wihf_w[t] + Wsf_w[t];
    }
    int b = t - HDIM * HDIM;
    if (b >= 0 && b < HDIM) {
        ws[        b] = Wih_b[b] + Wix_b[b] + bi[b];
        ws[ 128 +  b] = WTh_b[b] + WTx_b[b] + WTt_b[b] + bT[b];
        ws[ 256 +  b] = WLh_b[b] + WLx_b[b] + WLd_b[b] + bL[b];
        ws[ 384 +  b] = WzT_b[b] + WzL_b[b];
        ws[ 512 +  b] = Wuh_b[b] + Wux_b[b] + bu[b];
        ws[ 640 +  b] = Woh_b[b] + Wox_b[b] + Woz_b[b] + bo[b];
        ws[ 768 +  b] = Wr_b[b]  + br[b];
        ws[ 896 +  b] = Wihf_b[b] + Wsf_b[b] + Wrf_b[b] + bhf[b];
    }
    if (b == HDIM) {
        ws[1024] = Wy_b[0] + by[0];
    }
}

// ---------------------------------------------------------------------------
// Fused forward: each block handles 16 batch rows; 8 waves x 16 cols = H=128.
// All GEMMs use V_WMMA_F32_16X16X4_F32 (f32 precision), tiles staged in LDS.
// ---------------------------------------------------------------------------
__global__ __launch_bounds__(256) void stgn_fused(
    const float* __restrict__ X,      // (B, T, 66)
    const float* __restrict__ Wix,    // (64,128)
    const float* __restrict__ WTx,    // (64,128)
    const float* __restrict__ WLx,    // (64,128)
    const float* __restrict__ Wux,    // (64,128)
    const float* __restrict__ Wox,    // (64,128)
    const float* __restrict__ WTt,    // (1,128)
    const float* __restrict__ WLd,    // (1,128)
    const float* __restrict__ WzT,    // (1,128)
    const float* __restrict__ WzL,    // (1,128)
    const float* __restrict__ Woz,    // (128,128)
    const float* __restrict__ Wr,     // (128,128)
    const float* __restrict__ Wrf,    // (128,128)
    const float* __restrict__ WyW,    // (128,1)
    const float* __restrict__ ws,     // prep results
    float* __restrict__ out)          // (B,)
{
    __shared__ float xbuf[16 * DIN];    // x tile (16 x 64)
    __shared__ float dTs[16];
    __shared__ float dLs[16];
    __shared__ float zbuf[16 * HDIM];   // zeta tile, later reused for r
    __shared__ float hbuf[16 * HDIM];   // h_mid tile
    __shared__ float obuf[16 * HDIM];   // final h tile

    const int tid    = threadIdx.x;
    const int rowBase = blockIdx.x * 16;

    // ---- stage x, dT, dL of the LAST timestep into LDS -------------------
    for (int idx = tid; idx < 16 * FEAT; idx += 256) {
        int r  = idx / FEAT;
        int cc = idx - r * FEAT;
        float v = X[((size_t)(rowBase + r) * TLEN + (TLEN - 1)) * FEAT + cc];
        if (cc < DIN)       xbuf[r * DIN + cc] = v;
        else if (cc == DIN) dTs[r] = v;
        else                dLs[r] = v;
    }
    __syncthreads();

    const int lane    = tid & 31;
    const int nb      = tid >> 5;        // wave id = N-tile id (0..7)
    const int lanelow = lane & 15;       // M row (A) / N col (B, C/D)
    const int khalf   = lane >> 4;       // which K pair of the 4-K step
    const int col     = nb * 16 + lanelow;   // 0..127 column of H

    // per-lane constants (column-indexed)
    const float biE = ws[col],       bTE = ws[128 + col];
    const float bLE = ws[256 + col], bzE = ws[384 + col];
    const float buE = ws[512 + col], boE = ws[640 + col];
    const float brE = ws[768 + col], bhE = ws[896 + col];
    const float wtt = WTt[col], wld = WLd[col];
    const float wzt = WzT[col], wzl = WzL[col];

    // ---- 5 gate GEMMs over K=64 (shared A fragment) ----------------------
    v8f accI = {0,0,0,0,0,0,0,0};
    v8f accT = accI, accL = accI, accU = accI, accO = accI;
#pragma unroll
    for (int kk = 0; kk < DIN; kk += 4) {
        const int k0 = kk + 2 * khalf;
        v2f a; a.x = xbuf[lanelow * DIN + k0];
               a.y = xbuf[lanelow * DIN + k0 + 1];
        v2f b;
        b.x = Wix[(size_t)k0 * HDIM + col]; b.y = Wix[(size_t)(k0 + 1) * HDIM + col];
        accI = wmma_f32(a, b, accI);
        b.x = WTx[(size_t)k0 * HDIM + col]; b.y = WTx[(size_t)(k0 + 1) * HDIM + col];
        accT = wmma_f32(a, b, accT);
        b.x = WLx[(size_t)k0 * HDIM + col]; b.y = WLx[(size_t)(k0 + 1) * HDIM + col];
        accL = wmma_f32(a, b, accL);
        b.x = Wux[(size_t)k0 * HDIM + col]; b.y = Wux[(size_t)(k0 + 1) * HDIM + col];
        accU = wmma_f32(a, b, accU);
        b.x = Wox[(size_t)k0 * HDIM + col]; b.y = Wox[(size_t)(k0 + 1) * HDIM + col];
        accO = wmma_f32(a, b, accO);
    }

    // ---- gates + zeta (elementwise; C/D layout: row = e + 8*khalf) -------
    float tc[8];
#pragma unroll
    for (int e = 0; e < 8; ++e) {
        const int row = e + 8 * khalf;
        const float dT = dTs[row], dL = dLs[row];
        const float iG = sigm(accI[e] + biE);
        const float TG = sigm(accT[e] + dT * wtt + bTE);
        const float LG = sigm(accL[e] + dL * wld + bLE);
        const float ct = tanhf(accU[e] + buE);
        tc[e] = tanhf(iG * ct * TG * LG);           // tanh(c); (f*c0 == 0)
        zbuf[row * HDIM + col] = tanhf(dT * wzt + dL * wzl + bzE);  // zeta
    }
    __syncthreads();

    // ---- o2 = zeta @ Woz  (K=128) ---------------------------------------
    v8f acc2 = {0,0,0,0,0,0,0,0};
#pragma unroll
    for (int kk = 0; kk < HDIM; kk += 4) {
        const int k0 = kk + 2 * khalf;
        v2f a; a.x = zbuf[lanelow * HDIM + k0];
               a.y = zbuf[lanelow * HDIM + k0 + 1];
        v2f b; b.x = Woz[(size_t)k0 * HDIM + col];
               b.y = Woz[(size_t)(k0 + 1) * HDIM + col];
        acc2 = wmma_f32(a, b, acc2);
    }
    float hm[8];
#pragma unroll
    for (int e = 0; e < 8; ++e) {
        const int row = e + 8 * khalf;
        const float o = sigm(accO[e] + acc2[e] + boE);
        hm[e] = o * tc[e];                           // h_mid
        hbuf[row * HDIM + col] = hm[e];
    }
    __syncthreads();

    // ---- r = tanh(h_mid @ Wr + brEff)  (K=128) --------------------------
    v8f acc3 = {0,0,0,0,0,0,0,0};
#pragma unroll
    for (int kk = 0; kk < HDIM; kk += 4) {
        const int k0 = kk + 2 * khalf;
        v2f a; a.x = hbuf[lanelow * HDIM + k0];
               a.y = hbuf[lanelow * HDIM + k0 + 1];
        v2f b; b.x = Wr[(size_t)k0 * HDIM + col];
               b.y = Wr[(size_t)(k0 + 1) * HDIM + col];
        acc3 = wmma_f32(a, b, acc3);
    }
#pragma unroll
    for (int e = 0; e < 8; ++e) {
        const int row = e + 8 * khalf;
        zbuf[row * HDIM + col] = tanhf(acc3[e] + brE);   // r (reuse zbuf)
    }
    __syncthreads();

    // ---- h = tanh(h_mid @ (Wihf+Wsf) + r @ Wrf + bhEff)  (2x K=128) -----
    const float* Wfin = ws + 2048;
    v8f acc4 = {0,0,0,0,0,0,0,0};
#pragma unroll
    for (int kk = 0; kk < HDIM; kk += 4) {
        const int k0 = kk + 2 * khalf;
        v2f a; a.x = hbuf[lanelow * HDIM + k0];
               a.y = hbuf[lanelow * HDIM + k0 + 1];
        v2f b; b.x = Wfin[(size_t)k0 * HDIM + col];
               b.y = Wfin[(size_t)(k0 + 1) * HDIM + col];
        acc4 = wmma_f32(a, b, acc4);
    }
#pragma unroll
    for (int kk = 0; kk < HDIM; kk += 4) {
        const int k0 = kk + 2 * khalf;
        v2f a; a.x = zbuf[lanelow * HDIM + k0];
               a.y = zbuf[lanelow * HDIM + k0 + 1];
        v2f b; b.x = Wrf[(size_t)k0 * HDIM + col];
               b.y = Wrf[(size_t)(k0 + 1) * HDIM + col];
        acc4 = wmma_f32(a, b, acc4);
    }
#pragma unroll
    for (int e = 0; e < 8; ++e) {
        const int row = e + 8 * khalf;
        obuf[row * HDIM + col] = tanhf(acc4[e] + bhE);
    }
    __syncthreads();

    // ---- y = sigmoid(h @ Wy + wyEff): one lane per batch row -------------
    if (tid < 16) {
        float s = ws[1024];
#pragma unroll 8
        for (int k = 0; k < HDIM; ++k) s += obuf[tid * HDIM + k] * WyW[k];
        out[rowBase + tid] = sigm(s);
    }
}

// ---------------------------------------------------------------------------
// d_in layout: [0]=memory, [1]=X_seq, then params in dict-insertion order:
// sorted(_LINEARS) each contributing name_w, name_b; then Wy_w, Wy_b; then
// _EXTRA_BIASES; then by.  sorted(_LINEARS) =
//   WIAL, WIAT, WIh, WLdelta, WLh, WLx, WTh, WTt, WTx, WWh, Wfh, Wfx, Wih,
//   Wih_final, Wix, Woh, Wox, Woz, Wr, Wr_final, Ws_final, Wuh, Wux, WzL, WzT
// ---------------------------------------------------------------------------
extern "C" void kernel_launch(void* const* d_in, const int* in_sizes, int n_in,
                              void* d_out, int out_size, void* d_ws, size_t ws_size,
                              hipStream_t stream) {
    (void)in_sizes; (void)n_in; (void)out_size; (void)ws_size;
    const int P = 2;  // params base
    auto F = [&](int i) { return (const float*)d_in[i]; };

    const float* X          = F(1);
    // weights (name_w at P + 2*rank, name_b at P + 2*rank + 1)
    const float* WLd_w      = F(P + 6);   const float* WLd_b   = F(P + 7);
    const float* WLh_b      = F(P + 9);
    const float* WLx_w      = F(P + 10);  const float* WLx_b   = F(P + 11);
    const float* WTh_b      = F(P + 13);
    const float* WTt_w      = F(P + 14);  const float* WTt_b   = F(P + 15);
    const float* WTx_w      = F(P + 16);  const float* WTx_b   = F(P + 17);
    const float* Wih_b      = F(P + 25);
    const float* Wihf_w     = F(P + 26);  const float* Wihf_b  = F(P + 27);
    const float* Wix_w      = F(P + 28);  const float* Wix_b   = F(P + 29);
    const float* Woh_b      = F(P + 31);
    const float* Wox_w      = F(P + 32);  const float* Wox_b   = F(P + 33);
    const float* Woz_w      = F(P + 34);  const float* Woz_b   = F(P + 35);
    const float* Wr_w       = F(P + 36);  const float* Wr_b    = F(P + 37);
    const float* Wrf_w      = F(P + 38);  const float* Wrf_b   = F(P + 39);
    const float* Wsf_w      = F(P + 40);  const float* Wsf_b   = F(P + 41);
    const float* Wuh_b      = F(P + 43);
    const float* Wux_w      = F(P + 44);  const float* Wux_b   = F(P + 45);
    const float* WzL_w      = F(P + 46);  const float* WzL_b   = F(P + 47);
    const float* WzT_w      = F(P + 48);  const float* WzT_b   = F(P + 49);
    const float* Wy_w       = F(P + 50);  const float* Wy_b    = F(P + 51);
    // extra biases
    const float* bi  = F(P + 53);
    const float* bT  = F(P + 54);
    const float* bL  = F(P + 55);
    const float* bu  = F(P + 56);
    const float* bo  = F(P + 57);
    const float* br  = F(P + 59);
    const float* bhf = F(P + 60);
    const float* by  = F(P + 61);

    float* ws  = (float*)d_ws;
    float* out = (float*)d_out;

    // prep: 64 blocks for Wfin (128x128) + 1 block for biases
    stgn_prep<<<65, 256, 0, stream>>>(
        Wih_b, Wix_b, bi,
        WTh_b, WTx_b, WTt_b, bT,
        WLh_b, WLx_b, WLd_b, bL,
        WzT_b, WzL_b,
        Wuh_b, Wux_b, bu,
        Woh_b, Wox_b, Woz_b, bo,
        Wr_b, br,
        Wihf_b, Wsf_b, Wrf_b, bhf,
        Wy_b, by,
        Wihf_w, Wsf_w,
        ws);

    stgn_fused<<<BATCH / 16, 256, 0, stream>>>(
        X, Wix_w, WTx_w, WLx_w, Wux_w, Wox_w,
        WTt_w, WLd_w, WzT_w, WzL_w,
        Woz_w, Wr_w, Wrf_w, Wy_w,
        ws, out);
}